// DefectPredictionGNN_8950711845031
// MI455X (gfx1250) — compile-verified
//
#include <hip/hip_runtime.h>
#include <math.h>

#define N_NODES 50000
#define N_EDGES 800000
#define NB      64
#define HID     64
#define IN_CH   3
#define NDEF    6
#define BN_EPS  1e-5f

typedef float v2f __attribute__((ext_vector_type(2)));
typedef float v8f __attribute__((ext_vector_type(8)));

static inline unsigned gridFor(long long n, int blk = 256) {
    return (unsigned)((n + blk - 1) / blk);
}

// ---------------------------------------------------------------- degree
__global__ void k_deg_init(float* deg, int n) {
    int i = blockIdx.x * blockDim.x + threadIdx.x;
    if (i < n) deg[i] = 1.0f;                    // self-loop
}

__global__ void k_deg_accum(const int* __restrict__ dst, float* deg, int e) {
    int i = blockIdx.x * blockDim.x + threadIdx.x;
    if (i < e) atomicAdd(&deg[dst[i]], 1.0f);
}

__global__ void k_deg_to_dis(float* deg, int n) {
    int i = blockIdx.x * blockDim.x + threadIdx.x;
    if (i < n) deg[i] = rsqrtf(deg[i]);          // deg^{-1/2}
}

__global__ void k_enorm(const int* __restrict__ src, const int* __restrict__ dst,
                        const float* __restrict__ dis, float* __restrict__ enorm, int e) {
    int i = blockIdx.x * blockDim.x + threadIdx.x;
    if (i < e) enorm[i] = dis[src[i]] * dis[dst[i]];
}

// ---------------------------------------------------------------- layer-1 GEMM (K=3, VALU)
__global__ void k_gemm_in3(const float* __restrict__ x, const float* __restrict__ W,
                           float* __restrict__ m) {
    int i = blockIdx.x * blockDim.x + threadIdx.x;
    if (i >= N_NODES * HID) return;
    int node = i >> 6, f = i & 63;
    const float* xr = x + node * IN_CH;
    m[i] = fmaf(xr[0], W[0 * HID + f],
           fmaf(xr[1], W[1 * HID + f],
                xr[2] * W[2 * HID + f]));
}

// ---------------------------------------------------------------- 64x64 GEMM via fp32 WMMA
// m[N,64] = h[N,64] @ W[64,64]   (row-major). One wave per 16x16 output tile.
__global__ __launch_bounds__(32)
void k_gemm_wmma(const float* __restrict__ Hm, const float* __restrict__ W,
                 float* __restrict__ M, int n) {
    int tile = blockIdx.x;
    int tn = tile & 3;         // 4 column tiles (64/16)
    int tm = tile >> 2;        // row tile
    int lane = threadIdx.x;    // wave32
    int row  = tm * 16 + (lane & 15);
    int col  = tn * 16 + (lane & 15);
    int kb   = (lane >> 4) * 2;          // lanes 0-15 hold K+0/K+1, 16-31 hold K+2/K+3
    const float* hrow = Hm + (size_t)row * HID;

    v8f c = {};
    #pragma unroll
    for (int k0 = 0; k0 < HID; k0 += 4) {
        v2f a, b;
        a.x = hrow[k0 + kb];
        a.y = hrow[k0 + kb + 1];
        b.x = W[(k0 + kb) * HID + col];
        b.y = W[(k0 + kb + 1) * HID + col];
        // D = A(16x4,f32) * B(4x16,f32) + C ; full fp32
        c = __builtin_amdgcn_wmma_f32_16x16x4_f32(
                /*neg_a=*/false, a, /*neg_b=*/false, b,
                /*c_mod=*/(short)0, c, /*reuse_a=*/false, /*reuse_b=*/false);
    }

    int baseRow = tm * 16 + ((lane >> 4) ? 8 : 0);   // C/D: VGPR r -> M=r (+8 hi half)
    #pragma unroll
    for (int r = 0; r < 8; ++r) {
        int rr = baseRow + r;
        if (rr < n) M[(size_t)rr * HID + col] = c[r];
    }
}

// ---------------------------------------------------------------- agg = m*snorm + bias
__global__ void k_agg_init(const float* __restrict__ m, const float* __restrict__ dis,
                           const float* __restrict__ bias, float* __restrict__ agg) {
    int i = blockIdx.x * blockDim.x + threadIdx.x;
    if (i >= N_NODES * HID) return;
    int node = i >> 6, f = i & 63;
    float d = dis[node];
    agg[i] = fmaf(m[i], d * d, bias[f]);
}

// ---------------------------------------------------------------- edge scatter (L2-resident)
// 16 lanes per edge, each lane: one b128 gather + 4 float atomic adds.
__global__ void k_scatter(const int* __restrict__ src, const int* __restrict__ dst,
                          const float* __restrict__ enorm,
                          const float* __restrict__ m, float* __restrict__ agg) {
    int i = blockIdx.x * blockDim.x + threadIdx.x;
    if (i >= N_EDGES * 16) return;
    int e  = i >> 4;
    int f4 = (i & 15) << 2;
    int s = src[e], d = dst[e];
    float en = enorm[e];
    float4 v = *(const float4*)(m + (size_t)s * HID + f4);
    float* ad = agg + (size_t)d * HID + f4;
    atomicAdd(ad + 0, v.x * en);
    atomicAdd(ad + 1, v.y * en);
    atomicAdd(ad + 2, v.z * en);
    atomicAdd(ad + 3, v.w * en);
}

// ---------------------------------------------------------------- BN + ReLU
__global__ void k_bn_relu(const float* __restrict__ agg,
                          const float* __restrict__ g, const float* __restrict__ beta,
                          const float* __restrict__ mu, const float* __restrict__ var,
                          float* __restrict__ h) {
    int i = blockIdx.x * blockDim.x + threadIdx.x;
    if (i >= N_NODES * HID) return;
    int f = i & 63;
    float v = fmaf((agg[i] - mu[f]) * rsqrtf(var[f] + BN_EPS), g[f], beta[f]);
    h[i] = v > 0.0f ? v : 0.0f;
}

// ---------------------------------------------------------------- pooling
__global__ void k_pool_zero(float* gemb_cnt, int n) {
    int i = blockIdx.x * blockDim.x + threadIdx.x;
    if (i < n) gemb_cnt[i] = 0.0f;
}

__global__ void k_pool_accum(const int* __restrict__ batch, const float* __restrict__ h,
                             float* __restrict__ gemb, float* __restrict__ cnt) {
    int i = blockIdx.x * blockDim.x + threadIdx.x;
    if (i >= N_NODES * HID) return;
    int node = i >> 6, f = i & 63;
    int b = batch[node];
    atomicAdd(&gemb[b * HID + f], h[i]);
    if (f == 0) atomicAdd(&cnt[b], 1.0f);
}

// ---------------------------------------------------------------- heads: one graph per lane
__global__ __launch_bounds__(64)
void k_heads(const float* __restrict__ gemb, const float* __restrict__ cnt,
             const float* __restrict__ thW1, const float* __restrict__ thb1,
             const float* __restrict__ thW2, const float* __restrict__ thb2,
             const float* __restrict__ lhW1, const float* __restrict__ lhb1,
             const float* __restrict__ lhW2, const float* __restrict__ lhb2,
             const float* __restrict__ shW1, const float* __restrict__ shb1,
             const float* __restrict__ shW2, const float* __restrict__ shb2,
             float* __restrict__ out) {
    int b = threadIdx.x;
    if (b >= NB) return;
    float c = cnt[b];
    float inv = 1.0f / (c > 1.0f ? c : 1.0f);
    const float* gr = gemb + b * HID;

    float t[32];

    // type head: 64 -> 32 (relu) -> 6
    for (int j = 0; j < 32; ++j) {
        float s = thb1[j];
        for (int k = 0; k < HID; ++k) s = fmaf(gr[k] * inv, thW1[k * 32 + j], s);
        t[j] = s > 0.0f ? s : 0.0f;
    }
    for (int j = 0; j < NDEF; ++j) {
        float s = thb2[j];
        for (int k = 0; k < 32; ++k) s = fmaf(t[k], thW2[k * NDEF + j], s);
        out[b * NDEF + j] = s;
    }

    // location head: 64 -> 32 (relu) -> 2 (sigmoid)
    for (int j = 0; j < 32; ++j) {
        float s = lhb1[j];
        for (int k = 0; k < HID; ++k) s = fmaf(gr[k] * inv, lhW1[k * 32 + j], s);
        t[j] = s > 0.0f ? s : 0.0f;
    }
    for (int j = 0; j < 2; ++j) {
        float s = lhb2[j];
        for (int k = 0; k < 32; ++k) s = fmaf(t[k], lhW2[k * 2 + j], s);
        out[NB * NDEF + b * 2 + j] = 1.0f / (1.0f + expf(-s));
    }

    // severity head: 64 -> 16 (relu) -> 1 (sigmoid)
    for (int j = 0; j < 16; ++j) {
        float s = shb1[j];
        for (int k = 0; k < HID; ++k) s = fmaf(gr[k] * inv, shW1[k * 16 + j], s);
        t[j] = s > 0.0f ? s : 0.0f;
    }
    {
        float s = shb2[0];
        for (int k = 0; k < 16; ++k) s = fmaf(t[k], shW2[k], s);
        out[NB * NDEF + NB * 2 + b] = 1.0f / (1.0f + expf(-s));
    }
}

// ================================================================ launch
extern "C" void kernel_launch(void* const* d_in, const int* in_sizes, int n_in,
                              void* d_out, int out_size, void* d_ws, size_t ws_size,
                              hipStream_t stream) {
    const float* x    = (const float*)d_in[0];
    const int*   ei   = (const int*)d_in[1];
    const int*   srcp = ei;
    const int*   dstp = ei + N_EDGES;
    const int*   batch = (const int*)d_in[2];
    const float* W1 = (const float*)d_in[3];  const float* b1 = (const float*)d_in[4];
    const float* W2 = (const float*)d_in[5];  const float* b2 = (const float*)d_in[6];
    const float* W3 = (const float*)d_in[7];  const float* b3 = (const float*)d_in[8];
    const float* bnW[3][4];
    for (int l = 0; l < 3; ++l)
        for (int j = 0; j < 4; ++j)
            bnW[l][j] = (const float*)d_in[9 + l * 4 + j];     // g, b, m, v
    const float* thW1 = (const float*)d_in[21]; const float* thb1 = (const float*)d_in[22];
    const float* thW2 = (const float*)d_in[23]; const float* thb2 = (const float*)d_in[24];
    const float* lhW1 = (const float*)d_in[25]; const float* lhb1 = (const float*)d_in[26];
    const float* lhW2 = (const float*)d_in[27]; const float* lhb2 = (const float*)d_in[28];
    const float* shW1 = (const float*)d_in[29]; const float* shb1 = (const float*)d_in[30];
    const float* shW2 = (const float*)d_in[31]; const float* shb2 = (const float*)d_in[32];
    float* out = (float*)d_out;

    // workspace layout (floats)
    float* ws   = (float*)d_ws;
    float* dis  = ws;                              // N
    float* hbuf = dis + N_NODES;                   // N*H
    float* mbuf = hbuf + (size_t)N_NODES * HID;    // N*H
    float* abuf = mbuf + (size_t)N_NODES * HID;    // N*H
    float* gemb = abuf + (size_t)N_NODES * HID;    // NB*H
    float* cnt  = gemb + NB * HID;                 // NB
    float* enm  = cnt + NB;                        // E

    const long long NH = (long long)N_NODES * HID;
    const int TILES = (N_NODES / 16) * 4;          // 3125 * 4 wave-tiles

    // degree / normalization
    k_deg_init <<<gridFor(N_NODES), 256, 0, stream>>>(dis, N_NODES);
    k_deg_accum<<<gridFor(N_EDGES), 256, 0, stream>>>(dstp, dis, N_EDGES);
    k_deg_to_dis<<<gridFor(N_NODES), 256, 0, stream>>>(dis, N_NODES);
    k_enorm    <<<gridFor(N_EDGES), 256, 0, stream>>>(srcp, dstp, dis, enm, N_EDGES);

    // layer 1 (K=3 GEMM on VALU)
    k_gemm_in3 <<<gridFor(NH), 256, 0, stream>>>(x, W1, mbuf);
    k_agg_init <<<gridFor(NH), 256, 0, stream>>>(mbuf, dis, b1, abuf);
    k_scatter  <<<gridFor((long long)N_EDGES * 16), 256, 0, stream>>>(srcp, dstp, enm, mbuf, abuf);
    k_bn_relu  <<<gridFor(NH), 256, 0, stream>>>(abuf, bnW[0][0], bnW[0][1], bnW[0][2], bnW[0][3], hbuf);

    // layer 2 (fp32 WMMA GEMM)
    k_gemm_wmma<<<TILES, 32, 0, stream>>>(hbuf, W2, mbuf, N_NODES);
    k_agg_init <<<gridFor(NH), 256, 0, stream>>>(mbuf, dis, b2, abuf);
    k_scatter  <<<gridFor((long long)N_EDGES * 16), 256, 0, stream>>>(srcp, dstp, enm, mbuf, abuf);
    k_bn_relu  <<<gridFor(NH), 256, 0, stream>>>(abuf, bnW[1][0], bnW[1][1], bnW[1][2], bnW[1][3], hbuf);

    // layer 3 (fp32 WMMA GEMM)
    k_gemm_wmma<<<TILES, 32, 0, stream>>>(hbuf, W3, mbuf, N_NODES);
    k_agg_init <<<gridFor(NH), 256, 0, stream>>>(mbuf, dis, b3, abuf);
    k_scatter  <<<gridFor((long long)N_EDGES * 16), 256, 0, stream>>>(srcp, dstp, enm, mbuf, abuf);
    k_bn_relu  <<<gridFor(NH), 256, 0, stream>>>(abuf, bnW[2][0], bnW[2][1], bnW[2][2], bnW[2][3], hbuf);

    // pooling + heads
    k_pool_zero <<<gridFor(NB * HID + NB), 256, 0, stream>>>(gemb, NB * HID + NB);
    k_pool_accum<<<gridFor(NH), 256, 0, stream>>>(batch, hbuf, gemb, cnt);
    k_heads<<<1, 64, 0, stream>>>(gemb, cnt,
                                  thW1, thb1, thW2, thb2,
                                  lhW1, lhb1, lhW2, lhb2,
                                  shW1, shb1, shW2, shb2, out);
}